// ViTModel_15298673509057
// MI455X (gfx1250) — compile-verified
//
#include <hip/hip_runtime.h>

// ---------------------------------------------------------------------------
// ViT reference collapses: x drops out (pos-encoding aliasing), softmax over
// batch axis of identical scores -> uniform 1/B, so the whole model reduces to
// a single [64,16] -> [1024] vector pipeline + a 5-layer fp32 GEMV head
// (1024->512->256->128->64->10), then a broadcast of the 10-vector to 16384
// rows. One workgroup (32 wave32s) does the pipeline; WMMA_F32_16X16X4_F32
// executes the head GEMV chain on the matrix pipes.
// ---------------------------------------------------------------------------

typedef __attribute__((ext_vector_type(2))) float v2f;
typedef __attribute__((ext_vector_type(8))) float v8f;

#define NPATCH 64
#define PATCH  16
#define NTOT   1024          // 64*16
#define BATCH  16384
#define NOUT   10

__device__ __forceinline__ float block_layernorm(float v, int t, float* s,
                                                 float* red, float* red2,
                                                 float* stats) {
  __syncthreads();                 // make sure previous users of s are done
  s[t] = v;
  __syncthreads();
  if (t < 32) {
    float a = 0.f, b = 0.f;
    for (int i = 0; i < 32; ++i) {
      float x = s[t * 32 + i];
      a += x; b += x * x;
    }
    red[t] = a; red2[t] = b;
  }
  __syncthreads();
  if (t == 0) {
    float a = 0.f, b = 0.f;
    for (int i = 0; i < 32; ++i) { a += red[i]; b += red2[i]; }
    float m   = a * (1.0f / (float)NTOT);
    float var = b * (1.0f / (float)NTOT) - m * m;
    stats[0] = m;
    stats[1] = rsqrtf(var + 1e-5f);
  }
  __syncthreads();
  return (v - stats[0]) * stats[1];
}

__launch_bounds__(1024)
__global__ void vit_compute_kernel(
    const float* __restrict__ Wv,
    const float* __restrict__ w1, const float* __restrict__ b1,
    const float* __restrict__ w2, const float* __restrict__ b2,
    const float* __restrict__ hW1, const float* __restrict__ hb1,
    const float* __restrict__ hW2, const float* __restrict__ hb2,
    const float* __restrict__ hW3, const float* __restrict__ hb3,
    const float* __restrict__ hW4, const float* __restrict__ hb4,
    const float* __restrict__ hW5, const float* __restrict__ hb5,
    float* __restrict__ out10) {
  __shared__ float h0s[NTOT];
  __shared__ float h1s[NTOT];
  __shared__ float t0[NTOT];            // scratch (a1 / LN reductions)
  __shared__ float red[32];
  __shared__ float red2[32];
  __shared__ float zs[16];
  __shared__ float stats[2];
  __shared__ float fbuf[NTOT + 512 + 256 + 128 + 64 + 16];  // 2000 floats

  const int t = threadIdx.x;            // 0..1023
  const int p = t >> 4;                 // patch  0..63
  const int j = t & 15;                 // column 0..15

  // Prefetch the big head weight (2 MB) into cache hierarchy (global_prefetch_b8)
  for (int off = t * 32; off < 1024 * 512; off += 1024 * 32)
    __builtin_prefetch(hW1 + off, 0, 0);

  // ---- Stage 1: h0 = 2 * positional table (input x is dead) ----
  // div = p / 10000^(j/16*2) = p * exp(-j * 0.125 * ln(10000))
  float div = (float)p * __expf(-(float)j * 0.125f * 9.21034037198f);
  float pos = ((j & 1) == 0) ? __sinf(div) : __cosf(div);
  float h0  = 2.0f * pos;
  h0s[t] = h0;
  __syncthreads();

  // ---- Stage 2: z = colsum(h0 @ Wv) / B  (softmax over batch -> uniform) ----
  if (t < 16) {
    float cs = 0.f;
    for (int k = 0; k < NPATCH; ++k) cs += h0s[k * 16 + t];
    red[t] = cs;                        // colsum of h0
  }
  __syncthreads();
  if (t < 16) {
    float z = 0.f;
    for (int d = 0; d < 16; ++d) z += red[d] * Wv[d * 16 + t];
    zs[t] = z * (1.0f / (float)BATCH);
  }
  __syncthreads();

  // ---- Stage 3: h1 = LN(h0 + z) over all 1024 elements ----
  float h1 = block_layernorm(h0 + zs[j], t, t0, red, red2, stats);
  h1s[t] = h1;
  __syncthreads();

  // ---- Stage 4: 64 per-patch tiny MLPs (relu(W1)+relu(W2)) ----
  float a1 = b1[t];
  for (int i = 0; i < 16; ++i)
    a1 += h1s[p * 16 + i] * w1[p * 256 + i * 16 + j];
  a1 = fmaxf(a1, 0.f);
  __syncthreads();                      // t0 free (LN done)
  t0[t] = a1;
  __syncthreads();
  float a2 = b2[t];
  for (int i = 0; i < 16; ++i)
    a2 += t0[p * 16 + i] * w2[p * 256 + i * 16 + j];
  a2 = fmaxf(a2, 0.f);

  // ---- Stage 5: h2 = LN(a2 + h1); f0 = flatten(h2) ----
  float h2 = block_layernorm(a2 + h1, t, t0, red, red2, stats);
  fbuf[t] = h2;
  __syncthreads();

  // ---- Stage 6: head GEMV chain via V_WMMA_F32_16X16X4_F32 ----
  // A tile 16x4: row 0 carries the activation K-chunk (other rows zero),
  //   lanes 0-15 -> K={k,k+1}, lanes 16-31 -> K={k+2,k+3}  (ISA A layout)
  // B tile 4x16: lanes 0-15 -> rows k,k+1; lanes 16-31 -> rows k+2,k+3
  // D: row M=0 sits in acc.s0 of lanes 0-15 (N = lane).
  const int wave   = t >> 5;
  const int lane   = t & 31;
  const int laneN  = lane & 15;
  const int laneHi = lane >> 4;             // 0 or 1
  const float amask = (laneN == 0) ? 1.0f : 0.0f;

  const float* Ws[5] = {hW1, hW2, hW3, hW4, hW5};
  const float* Bs[5] = {hb1, hb2, hb3, hb4, hb5};
  const int fins[5]  = {1024, 512, 256, 128, 64};
  const int fouts[5] = {512, 256, 128, 64, NOUT};

  int inOff = 0;
  for (int l = 0; l < 5; ++l) {
    const int fin    = fins[l];
    const int fout   = fouts[l];
    const int outOff = inOff + fin;
    const int ntiles = (fout + 15) >> 4;
    const float* W  = Ws[l];
    const float* Bb = Bs[l];

    for (int tile = wave; tile < ntiles; tile += 32) {
      const int n  = tile * 16 + laneN;
      const int nc = (n < fout) ? n : (fout - 1);    // clamp (no OOB)
      const float bmask = (n < fout) ? 1.0f : 0.0f;  // mask (no divergence)
      v8f acc = {};
      for (int kb = 0; kb < fin; kb += 4) {
        const int ka = kb + laneHi * 2;
        v2f a, b;
        a.x = amask * fbuf[inOff + ka];
        a.y = amask * fbuf[inOff + ka + 1];
        b.x = bmask * W[(ka + 0) * fout + nc];
        b.y = bmask * W[(ka + 1) * fout + nc];
        acc = __builtin_amdgcn_wmma_f32_16x16x4_f32(
            /*neg_a=*/false, a, /*neg_b=*/false, b,
            /*c_mod=*/(short)0, acc, /*reuse_a=*/false, /*reuse_b=*/false);
      }
      if (laneHi == 0 && n < fout)
        fbuf[outOff + n] = acc.s0 + Bb[n];
    }
    __syncthreads();
    inOff = outOff;
  }

  // ---- Stage 7: emit the 10-vector to workspace ----
  if (t < NOUT) out10[t] = fbuf[inOff + t];
}

__global__ void vit_broadcast_kernel(const float* __restrict__ out10,
                                     float* __restrict__ out, int total) {
  __shared__ float v[NOUT];
  if (threadIdx.x < NOUT) v[threadIdx.x] = out10[threadIdx.x];
  __syncthreads();
  int i = blockIdx.x * blockDim.x + threadIdx.x;
  if (i < total) out[i] = v[i % NOUT];
}

extern "C" void kernel_launch(void* const* d_in, const int* in_sizes, int n_in,
                              void* d_out, int out_size, void* d_ws, size_t ws_size,
                              hipStream_t stream) {
  // d_in order: 0:x 1:Wq 2:Wk 3:Wv 4:mlp_w1 5:mlp_b1 6:mlp_w2 7:mlp_b2
  //             8:hW1 9:hb1 10:hW2 11:hb2 12:hW3 13:hb3 14:hW4 15:hb4 16:hW5 17:hb5
  // (x, Wq, Wk are mathematically dead in the reference.)
  const float* Wv  = (const float*)d_in[3];
  const float* w1  = (const float*)d_in[4];
  const float* b1  = (const float*)d_in[5];
  const float* w2  = (const float*)d_in[6];
  const float* b2  = (const float*)d_in[7];
  const float* hW1 = (const float*)d_in[8];
  const float* hb1 = (const float*)d_in[9];
  const float* hW2 = (const float*)d_in[10];
  const float* hb2 = (const float*)d_in[11];
  const float* hW3 = (const float*)d_in[12];
  const float* hb3 = (const float*)d_in[13];
  const float* hW4 = (const float*)d_in[14];
  const float* hb4 = (const float*)d_in[15];
  const float* hW5 = (const float*)d_in[16];
  const float* hb5 = (const float*)d_in[17];

  float* scratch10 = (float*)d_ws;         // 10 floats
  float* out       = (float*)d_out;        // 16384 x 10 fp32

  vit_compute_kernel<<<1, 1024, 0, stream>>>(
      Wv, w1, b1, w2, b2,
      hW1, hb1, hW2, hb2, hW3, hb3, hW4, hb4, hW5, hb5,
      scratch10);

  const int total = out_size;              // 163840
  vit_broadcast_kernel<<<(total + 255) / 256, 256, 0, stream>>>(
      scratch10, out, total);
}